// Hybrid_38663295599269
// MI455X (gfx1250) — compile-verified
//
#include <hip/hip_runtime.h>
#include <hip/hip_bf16.h>

// ---------------------------------------------------------------------------
// Fused 8-layer MLP ensemble for MI455X (gfx1250, wave32, WMMA).
//   big path : y = y @ Wbig[l]  + Bbig[l]        (W stored K-major: [f_in, f_out])
//   lil path : y = y @ Wlil[l,m]^T + Blil[l,m]   (W stored N-major: [g, f] -> B[k,n]=W[n,k])
// Both paths share one kernel template parameterized by the B-matrix element
// strides SK (along K) and SN (along N) so offsets fold into immediates.
// fp32 WMMA (V_WMMA_F32_16X16X4_F32): exact reference numerics; the problem is
// HBM/L2 bound (~200 MB moved vs 137 GFLOP), so low-precision buys nothing.
// Input tile staging uses GLOBAL_LOAD_ASYNC_TO_LDS_B128 (ASYNCcnt path).
// ---------------------------------------------------------------------------

#define FDIM      512
#define NLAYERS   8
#define MTILE     64            // rows per block
#define LDS_STRIDE 516          // floats per padded LDS row (516 % 64 banks = 4; 2064B rows keep 16B align)

typedef __attribute__((ext_vector_type(2))) float v2f;
typedef __attribute__((ext_vector_type(4))) float v4f;
typedef __attribute__((ext_vector_type(8))) float v8f;

template <int SK, int SN>
__launch_bounds__(256, 1)
__global__ void mlp8_fused(const float* __restrict__ X,
                           const float* __restrict__ Wbase,
                           const float* __restrict__ Bbase,
                           float* __restrict__ Out,
                           int row0_base,          // first global row of block 0
                           int mshift,             // model = blockIdx.x >> mshift
                           int wLayerStride,       // floats between layers of one model
                           int bLayerStride)       // floats between layer biases
{
    __shared__ float act[MTILE * LDS_STRIDE];

    const int tid  = threadIdx.x;
    const int lane = tid & 31;
    const int wave = tid >> 5;
    const int rb   = wave & 3;                  // 16-row block within tile
    const int ch   = wave >> 2;                 // 256-col half
    const int m    = (int)(blockIdx.x >> mshift);
    const int row0 = row0_base + (int)blockIdx.x * MTILE;

    const float* W0 = Wbase + (size_t)m * FDIM * FDIM;
    const float* B0 = Bbase + (size_t)m * FDIM;

    // ---- stage input tile X[row0 .. row0+MTILE) into LDS ----
    // 64*512/4 = 8192 float4s, 32 per thread.
#if defined(__gfx1250__)
    // Async DMA memory -> LDS (no VGPR round trip), tracked by ASYNCcnt.
    for (int i = 0; i < 32; ++i) {
        int f = i * 256 + tid;
        int r = f >> 7;               // / 128 float4s per row
        int c = (f & 127) << 2;
        unsigned ldsAddr = (unsigned)(size_t)(void*)&act[r * LDS_STRIDE + c];
        const float* g = X + (size_t)(row0 + r) * FDIM + c;
        asm volatile("global_load_async_to_lds_b128 %0, %1, off"
                     :: "v"(ldsAddr), "v"(g) : "memory");
    }
    asm volatile("s_wait_asynccnt 0x0" ::: "memory");
#else
    #pragma unroll 4
    for (int i = 0; i < 32; ++i) {
        int f = i * 256 + tid;
        int r = f >> 7;
        int c = (f & 127) << 2;
        v4f v = *(const v4f*)(X + (size_t)(row0 + r) * FDIM + c);
        *(v4f*)(&act[r * LDS_STRIDE + c]) = v;
    }
#endif
    __syncthreads();

    // Per-lane fragment coordinates (ISA 7.12.2 layouts, wave32):
    //  A 16x4 f32 : lane holds row (lane&15), K = kk,kk+1 with kk = 2*(lane>>4)
    //  B 4x16 f32 : lane holds col (lane&15), K = kk,kk+1
    //  C/D 16x16  : vgpr v, lane -> row v + 8*(lane>>4), col lane&15
    const int kk = (lane >> 4) << 1;
    const int nC = ch * 256 + (lane & 15);          // base output column (ct = 0)
    const float* aRow = act + (size_t)(rb * 16 + (lane & 15)) * LDS_STRIDE + kk;

    for (int l = 0; l < NLAYERS; ++l) {
        const float* Wl = W0 + (size_t)l * wLayerStride;
        const float* Bl = B0 + (size_t)l * bLayerStride;

        // init accumulators with bias (exact y@W + b)
        v8f acc[16];
        #pragma unroll
        for (int ct = 0; ct < 16; ++ct) {
            float bv = Bl[nC + ct * 16];
            #pragma unroll
            for (int v = 0; v < 8; ++v) acc[ct][v] = bv;
        }

        // warm next layer's weights in L2 (global_prefetch_b8)
        if (l + 1 < NLAYERS) {
            const float* wn = W0 + (size_t)(l + 1) * wLayerStride;
            __builtin_prefetch(wn + (size_t)tid * 256, 0, 1);
            __builtin_prefetch(wn + (size_t)tid * 256 + 65536, 0, 1);
        }

        // K loop: 128 steps of K=4, 16 WMMAs per step
        const float* wk = Wl + (size_t)kk * SK + (size_t)nC * SN;   // per-lane base
        for (int k0 = 0; k0 < FDIM; k0 += 4) {
            v2f a = *(const v2f*)(aRow + k0);
            #pragma unroll
            for (int ct = 0; ct < 16; ++ct) {
                v2f b;
                b.x = wk[(size_t)ct * 16 * SN];        // B[k0+kk  ][nC+16ct]
                b.y = wk[(size_t)ct * 16 * SN + SK];   // B[k0+kk+1][nC+16ct]
                acc[ct] = __builtin_amdgcn_wmma_f32_16x16x4_f32(
                    false, a, false, b, (short)0, acc[ct], false, false);
            }
            wk += (size_t)4 * SK;
        }

        __syncthreads();   // every wave finished reading this layer's activations

        const int rOut = rb * 16 + 8 * (lane >> 4);
        if (l == NLAYERS - 1) {
            // final layer: straight to global output
            #pragma unroll
            for (int ct = 0; ct < 16; ++ct) {
                const int col = nC + ct * 16;
                #pragma unroll
                for (int v = 0; v < 8; ++v)
                    Out[(size_t)(row0 + rOut + v) * FDIM + col] = acc[ct][v];
            }
        } else {
            // write back to LDS for the next layer
            #pragma unroll
            for (int ct = 0; ct < 16; ++ct) {
                const int col = nC + ct * 16;
                #pragma unroll
                for (int v = 0; v < 8; ++v)
                    act[(size_t)(rOut + v) * LDS_STRIDE + col] = acc[ct][v];
            }
            __syncthreads();   // next layer sees completed activations
        }
    }
}

extern "C" void kernel_launch(void* const* d_in, const int* in_sizes, int n_in,
                              void* d_out, int out_size, void* d_ws, size_t ws_size,
                              hipStream_t stream) {
    const float* x    = (const float*)d_in[0];   // (32768, 512)
    const float* Wbig = (const float*)d_in[1];   // (8, 512, 512)
    const float* Bbig = (const float*)d_in[2];   // (8, 512)
    const float* Wlil = (const float*)d_in[3];   // (8, 8, 512, 512)
    const float* Blil = (const float*)d_in[4];   // (8, 8, 512)
    float* out = (float*)d_out;                  // (32768, 512)

    const int split = 16384;                     // SPLIT
    const int nLilModels = 8;                    // N_MODELS - 1
    const int nLil = 2048;                       // rows per lil model

    // big: 16384 rows -> 256 blocks; B element (k,n) at W[k*512 + n]
    mlp8_fused<FDIM, 1><<<split / MTILE, 256, 0, stream>>>(
        x, Wbig, Bbig, out,
        /*row0_base=*/0, /*mshift=*/30,
        /*wLayerStride=*/FDIM * FDIM, /*bLayerStride=*/FDIM);

    // lil: 8 models * 2048 rows -> 256 blocks (32 per model);
    // B element (k,n) at W[n*512 + k]  (einsum 'mgf': W is [g,f] = [N,K])
    mlp8_fused<1, FDIM><<<(nLilModels * nLil) / MTILE, 256, 0, stream>>>(
        x, Wlil, Blil, out,
        /*row0_base=*/split, /*mshift=*/5,
        /*wLayerStride=*/nLilModels * FDIM * FDIM, /*bLayerStride=*/nLilModels * FDIM);

    (void)in_sizes; (void)n_in; (void)out_size; (void)d_ws; (void)ws_size;
}